// DeepAE_MODEL_20255065768607
// MI455X (gfx1250) — compile-verified
//
#include <hip/hip_runtime.h>

typedef __attribute__((ext_vector_type(2))) float v2f;
typedef __attribute__((ext_vector_type(8))) float v8f;

// -------------------- graph normalization --------------------

__global__ void init_deg_kernel(float* __restrict__ deg, int n) {
    int i = blockIdx.x * blockDim.x + threadIdx.x;
    if (i < n) deg[i] = 1.0f;   // self-loop contributes 1 to every node's degree
}

__global__ void deg_accum_kernel(float* __restrict__ deg,
                                 const int* __restrict__ col, int e) {
    int i = blockIdx.x * blockDim.x + threadIdx.x;
    if (i < e) atomicAdd(&deg[col[i]], 1.0f);
}

__global__ void dinv_kernel(float* __restrict__ deg, int n) {
    int i = blockIdx.x * blockDim.x + threadIdx.x;
    if (i < n) deg[i] = __frsqrt_rn(deg[i]);   // deg >= 1 always (self loops)
}

__global__ void norm_kernel(const float* __restrict__ dinv,
                            const int* __restrict__ row,
                            const int* __restrict__ col,
                            float* __restrict__ norm, int e) {
    int i = blockIdx.x * blockDim.x + threadIdx.x;
    if (i < e) norm[i] = dinv[row[i]] * dinv[col[i]];
}

// -------------------- dense GEMM via V_WMMA_F32_16X16X4_F32 --------------------
// One wave computes a 16-row band x FOUT cols. A is [nrows,K] row-major,
// W is [K,FOUT] row-major, Out is [nrows,FOUT] row-major.
// A frag (16x4 f32): lane<16 holds {K=k0,k0+1} of row l; lane>=16 holds {k0+2,k0+3}.
// B frag (4x16 f32): lane<16 VGPR0/1 = W[k0][c0+l], W[k0+1][c0+l]; lanes 16-31 = K+2/K+3.
// D (16x16 f32): lane group h, vgpr j -> D[8*h + j][c0 + l].

template<int K, int FOUT>
__global__ void gemm_wmma_kernel(const float* __restrict__ A,
                                 const float* __restrict__ W,
                                 float* __restrict__ Out, int nrows) {
    constexpr int T = FOUT / 16;
    int wave = (blockIdx.x * blockDim.x + threadIdx.x) >> 5;
    int lane = threadIdx.x & 31;
    int row0 = wave * 16;
    if (row0 >= nrows) return;                 // wave-uniform: EXEC stays all-ones
    int half = lane >> 4;
    int l    = lane & 15;

    v8f zero = {};
    v8f acc[T];
#pragma unroll
    for (int t = 0; t < T; ++t) acc[t] = zero;

    const float* arow = A + (size_t)(row0 + l) * K;
#pragma unroll 4
    for (int k0 = 0; k0 < K; k0 += 4) {
        int ka = k0 + 2 * half;
        v2f a = *reinterpret_cast<const v2f*>(arow + ka);   // contiguous, 8B aligned
        const float* w0 = W + (size_t)ka * FOUT + l;
        const float* w1 = w0 + FOUT;
#pragma unroll
        for (int t = 0; t < T; ++t) {
            v2f b;
            b.x = w0[t * 16];
            b.y = w1[t * 16];
            acc[t] = __builtin_amdgcn_wmma_f32_16x16x4_f32(
                /*neg_a=*/false, a, /*neg_b=*/false, b,
                /*c_mod=*/(short)0, acc[t],
                /*reuse_a=*/false, /*reuse_b=*/false);
        }
    }

#pragma unroll
    for (int t = 0; t < T; ++t) {
#pragma unroll
        for (int j = 0; j < 8; ++j) {
            int m = 8 * half + j;
            Out[(size_t)(row0 + m) * FOUT + t * 16 + l] = acc[t][j];
        }
    }
}

// -------------------- edge-parallel scatter (one wave32 per edge) --------------------

template<int F>
__global__ void scatter_kernel(const float* __restrict__ HW,
                               float* __restrict__ AGG,
                               const int* __restrict__ row,
                               const int* __restrict__ col,
                               const float* __restrict__ norm, int e) {
    int wave = (blockIdx.x * blockDim.x + threadIdx.x) >> 5;
    int lane = threadIdx.x & 31;
    if (wave >= e) return;
    int r = row[wave];
    int c = col[wave];
    float w = norm[wave];
    constexpr int PER = F / 32;                // 2 (F=64) or 4 (F=128) floats per lane
    const float* src = HW + (size_t)r * F + lane * PER;
    float*       dst = AGG + (size_t)c * F + lane * PER;
#pragma unroll
    for (int i = 0; i < PER; ++i)
        atomicAdd(dst + i, src[i] * w);
}

// -------------------- combine: (1-c)*hw + c*(agg + dinv^2*hw) + bias (+relu) ------

template<bool RELU>
__global__ void combine_kernel(const float* __restrict__ HW,
                               const float* __restrict__ AGG,
                               const float* __restrict__ dinv,
                               const float* __restrict__ bias,
                               float* __restrict__ Out,
                               int total, int F, float a1, float coef) {
    int idx = blockIdx.x * blockDim.x + threadIdx.x;
    if (idx >= total) return;
    int i = idx / F;
    int f = idx - i * F;
    float dv = dinv[i];
    float hw = HW[idx];
    float v  = a1 * hw + coef * (AGG[idx] + dv * dv * hw) + bias[f];
    if (RELU) v = fmaxf(v, 0.0f);
    Out[idx] = v;
}

// -------------------- host-side layer driver --------------------

static void run_layer(const float* h, const float* W, const float* bias,
                      float* hw, float* agg, float* out,
                      int K, int F, float coef, bool relu,
                      const int* row, const int* col, const float* norm,
                      const float* dinv, int N, int E, hipStream_t stream) {
    int bands      = N / 16;                       // N = 100000 -> 6250 full bands
    int gemmBlocks = (bands + 7) / 8;              // 8 waves (256 threads) per block
    if (K == 128 && F == 64)
        gemm_wmma_kernel<128, 64><<<gemmBlocks, 256, 0, stream>>>(h, W, hw, N);
    else if (K == 64 && F == 64)
        gemm_wmma_kernel<64, 64><<<gemmBlocks, 256, 0, stream>>>(h, W, hw, N);
    else
        gemm_wmma_kernel<64, 128><<<gemmBlocks, 256, 0, stream>>>(h, W, hw, N);

    hipMemsetAsync(agg, 0, (size_t)N * F * sizeof(float), stream);

    int sBlocks = (E + 7) / 8;                     // one wave per edge
    if (F == 64)
        scatter_kernel<64><<<sBlocks, 256, 0, stream>>>(hw, agg, row, col, norm, E);
    else
        scatter_kernel<128><<<sBlocks, 256, 0, stream>>>(hw, agg, row, col, norm, E);

    int total   = N * F;
    int cBlocks = (total + 255) / 256;
    float a1 = 1.0f - coef;
    if (relu)
        combine_kernel<true ><<<cBlocks, 256, 0, stream>>>(hw, agg, dinv, bias, out, total, F, a1, coef);
    else
        combine_kernel<false><<<cBlocks, 256, 0, stream>>>(hw, agg, dinv, bias, out, total, F, a1, coef);
}

extern "C" void kernel_launch(void* const* d_in, const int* in_sizes, int n_in,
                              void* d_out, int out_size, void* d_ws, size_t ws_size,
                              hipStream_t stream) {
    const float* x   = (const float*)d_in[0];
    const int*   ei  = (const int*)  d_in[1];
    const float* we0 = (const float*)d_in[2];  const float* be0 = (const float*)d_in[3];
    const float* we1 = (const float*)d_in[4];  const float* be1 = (const float*)d_in[5];
    const float* we2 = (const float*)d_in[6];  const float* be2 = (const float*)d_in[7];
    const float* wd0 = (const float*)d_in[8];  const float* bd0 = (const float*)d_in[9];
    const float* wd1 = (const float*)d_in[10]; const float* bd1 = (const float*)d_in[11];
    const float* wd2 = (const float*)d_in[12]; const float* bd2 = (const float*)d_in[13];
    float* out = (float*)d_out;

    const int N = in_sizes[0] / 128;     // 100000
    const int E = in_sizes[1] / 2;       // 1600000
    const int* row = ei;                 // edge_index[0] = source nodes
    const int* col = ei + E;             // edge_index[1] = target nodes

    // ---- workspace layout (floats): dinv[N] | norm[E] | 3 x [N*128] ----
    float* ws    = (float*)d_ws;
    float* dinv  = ws;
    float* norm  = dinv + N;
    size_t bufsz = (size_t)N * 128;
    float* buf0  = norm + E;
    float* buf1  = buf0 + bufsz;
    float* buf2  = buf1 + bufsz;

    // ---- gcn_norm: deg (with self loop), dinv = rsqrt(deg), per-edge norm ----
    init_deg_kernel <<<(N + 255) / 256, 256, 0, stream>>>(dinv, N);
    deg_accum_kernel<<<(E + 255) / 256, 256, 0, stream>>>(dinv, col, E);
    dinv_kernel     <<<(N + 255) / 256, 256, 0, stream>>>(dinv, N);
    norm_kernel     <<<(E + 255) / 256, 256, 0, stream>>>(dinv, row, col, norm, E);

    float* z       = out;                     // [N, 64]
    float* dec_out = out + (size_t)N * 64;    // [N, 128]

    // ---- encoder (coef = 1.0) ----
    run_layer(x,    we0, be0, buf0, buf1, buf1, 128, 64, 1.0f, true,  row, col, norm, dinv, N, E, stream);
    run_layer(buf1, we1, be1, buf0, buf2, buf2,  64, 64, 1.0f, true,  row, col, norm, dinv, N, E, stream);
    run_layer(buf2, we2, be2, buf0, buf1, z,     64, 64, 1.0f, false, row, col, norm, dinv, N, E, stream);

    // ---- decoder (SharpenGCN, coef = 0.5) ----
    run_layer(z,    wd0, bd0, buf0, buf1, buf1,  64, 64, 0.5f, true,  row, col, norm, dinv, N, E, stream);
    run_layer(buf1, wd1, bd1, buf0, buf2, buf2,  64, 64, 0.5f, true,  row, col, norm, dinv, N, E, stream);
    run_layer(buf2, wd2, bd2, buf0, buf1, dec_out, 64, 128, 0.5f, false, row, col, norm, dinv, N, E, stream);
}